// CES_CE_52613349376066
// MI455X (gfx1250) — compile-verified
//
#include <hip/hip_runtime.h>
#include <cstdint>
#include <cstddef>

#define AS1 __attribute__((address_space(1)))
#define AS3 __attribute__((address_space(3)))

typedef int v4i __attribute__((ext_vector_type(4)));

// ---------------------------------------------------------------------------
// CDNA5 async global -> LDS copy (ASYNCcnt-tracked), 16 bytes per lane.
// Builtin if the toolchain declares it, else raw gfx1250 inline asm.
// Builtin signature (from diagnostics): (v4i AS1*, v4i AS3*, imm int, imm int)
// ---------------------------------------------------------------------------
__device__ __forceinline__ void async_ld16(void* lds_p, const void* g_p) {
#if __has_builtin(__builtin_amdgcn_global_load_async_to_lds_b128)
  __builtin_amdgcn_global_load_async_to_lds_b128(
      (AS1 v4i*)g_p, (AS3 v4i*)lds_p, 0, 0);
#else
  unsigned lds_addr = (unsigned)(size_t)(AS3 char*)lds_p;
  asm volatile("global_load_async_to_lds_b128 %0, %1, off"
               :
               : "v"(lds_addr), "v"(g_p)
               : "memory");
#endif
}

template <int N>
__device__ __forceinline__ void wait_async() {
#if __has_builtin(__builtin_amdgcn_s_wait_asynccnt)
  __builtin_amdgcn_s_wait_asynccnt(N);
#else
  asm volatile("s_wait_asynccnt %0" : : "i"(N) : "memory");
#endif
}

// ---------------------------------------------------------------------------
// wave32 + LDS block reductions
// ---------------------------------------------------------------------------
__device__ __forceinline__ float wred_max(float v) {
#pragma unroll
  for (int o = 16; o > 0; o >>= 1) v = fmaxf(v, __shfl_xor(v, o, 32));
  return v;
}
__device__ __forceinline__ float wred_sum(float v) {
#pragma unroll
  for (int o = 16; o > 0; o >>= 1) v += __shfl_xor(v, o, 32);
  return v;
}

__device__ float block_max(float v) {
  __shared__ float red[8];
  const int lane = threadIdx.x & 31, w = threadIdx.x >> 5;
  v = wred_max(v);
  if (lane == 0) red[w] = v;
  __syncthreads();
  if (w == 0) {
    float t = (lane < (int)((blockDim.x + 31) >> 5)) ? red[lane] : -3.402823466e38f;
    t = wred_max(t);
    if (lane == 0) red[0] = t;
  }
  __syncthreads();
  float r = red[0];
  __syncthreads();
  return r;
}

__device__ float block_sum(float v) {
  __shared__ float red[8];
  const int lane = threadIdx.x & 31, w = threadIdx.x >> 5;
  v = wred_sum(v);
  if (lane == 0) red[w] = v;
  __syncthreads();
  if (w == 0) {
    float t = (lane < (int)((blockDim.x + 31) >> 5)) ? red[lane] : 0.0f;
    t = wred_sum(t);
    if (lane == 0) red[0] = t;
  }
  __syncthreads();
  float r = red[0];
  __syncthreads();
  return r;
}

// ---------------------------------------------------------------------------
// Problem constants
// ---------------------------------------------------------------------------
namespace {
constexpr int   TPB    = 256;
constexpr int   C_FIX  = 10000;
constexpr int   V4     = C_FIX / 4;                 // 2500 float4 per row
constexpr int   CHUNKS = (V4 + TPB - 1) / TPB;      // 10 async issues / thread
constexpr int   BUF_F4 = CHUNKS * TPB;              // 2560 float4 = 40960 B / buffer
constexpr float A_COEF = -4.0f;
constexpr float a_coef = 1.0f;
constexpr float b_coef = 1.0f;
}  // namespace

// ---------------------------------------------------------------------------
// Fast path: double-buffered async row staging in LDS, two passes over LDS.
// HBM traffic = exactly one read per element (bandwidth-optimal: ~655 MB,
// floor ~28 us at 23.3 TB/s).
// ---------------------------------------------------------------------------
__global__ __launch_bounds__(TPB) void ces_rows_fast(
    const float* __restrict__ x, const int* __restrict__ tgt,
    float* __restrict__ ce_arr, float* __restrict__ pt_arr, int Brows) {
  extern __shared__ float4 lbuf[];  // 2 * BUF_F4 float4
  const int tid = threadIdx.x;

  auto issue_row = [&](int r, int which) {
    const char* gbase = (const char*)x + (size_t)r * (size_t)C_FIX * 4u;
    char* lbase = (char*)lbuf + (size_t)which * (size_t)BUF_F4 * 16u;
#pragma unroll
    for (int c = 0; c < CHUNKS; ++c) {
      const int i4  = c * TPB + tid;
      const int i4c = (i4 < V4) ? i4 : (V4 - 1);  // clamp tail: uniform issue count
      async_ld16(lbase + (size_t)i4 * 16u, gbase + (size_t)i4c * 16u);
    }
  };

  int row = blockIdx.x;
  const int rstep = gridDim.x;
  if (row < Brows) issue_row(row, 0);
  int cur = 0;

  for (; row < Brows; row += rstep) {
    const int nrow = row + rstep;
    if (nrow < Brows) {
      issue_row(nrow, cur ^ 1);   // overlap next row's DMA with this row's math
      wait_async<CHUNKS>();       // ASYNCcnt in-order: current buffer complete
    } else {
      wait_async<0>();
    }
    __syncthreads();

    const float4* buf = lbuf + (size_t)cur * BUF_F4;

    // pass 1: row max (LDS)
    float m = -3.402823466e38f;
    for (int j = tid; j < V4; j += TPB) {
      const float4 v = buf[j];
      m = fmaxf(fmaxf(m, v.x), fmaxf(v.y, fmaxf(v.z, v.w)));
    }
    m = block_max(m);

    // pass 2: sum exp(x - m) (LDS again — free second read)
    float s = 0.0f;
    for (int j = tid; j < V4; j += TPB) {
      const float4 v = buf[j];
      s += __expf(v.x - m) + __expf(v.y - m) + __expf(v.z - m) + __expf(v.w - m);
    }
    s = block_sum(s);

    if (tid == 0) {
      const float xt = ((const float*)buf)[tgt[row]];  // gather from LDS-resident row
      const float lp = xt - m - __logf(s);             // logp_t
      ce_arr[row] = -lp;
      pt_arr[row] = __expf(lp);                        // p_t
    }
    __syncthreads();  // all reads of buf[cur] done before it is DMA'd into next iter
    cur ^= 1;
  }
}

// ---------------------------------------------------------------------------
// Generic fallback (any C), direct global loads.
// ---------------------------------------------------------------------------
__global__ __launch_bounds__(256) void ces_rows_generic(
    const float* __restrict__ x, const int* __restrict__ tgt,
    float* __restrict__ ce_arr, float* __restrict__ pt_arr, int Brows, int C) {
  for (int row = blockIdx.x; row < Brows; row += gridDim.x) {
    const float* r = x + (size_t)row * (size_t)C;
    float m = -3.402823466e38f;
    for (int j = threadIdx.x; j < C; j += blockDim.x) m = fmaxf(m, r[j]);
    m = block_max(m);
    float s = 0.0f;
    for (int j = threadIdx.x; j < C; j += blockDim.x) s += __expf(r[j] - m);
    s = block_sum(s);
    if (threadIdx.x == 0) {
      const float lp = r[tgt[row]] - m - __logf(s);
      ce_arr[row] = -lp;
      pt_arr[row] = __expf(lp);
    }
    __syncthreads();
  }
}

// ---------------------------------------------------------------------------
// Deterministic final reduction -> scalar
// ---------------------------------------------------------------------------
__global__ __launch_bounds__(256) void ces_finalize(
    const float* __restrict__ ce_arr, const float* __restrict__ pt_arr,
    float* __restrict__ out, int Brows) {
  float sc = 0.0f, sp = 0.0f;
  for (int i = threadIdx.x; i < Brows; i += blockDim.x) {
    sc += ce_arr[i];
    sp += pt_arr[i];
  }
  sc = block_sum(sc);
  sp = block_sum(sp);
  if (threadIdx.x == 0) {
    const float Bf  = (float)Brows;
    const float ce  = sc / Bf;                       // -(1/B) sum logp_t
    const float ces = (sp - Bf) * (A_COEF / Bf);     // (sum p_t - sum p) * A/B
    out[0] = a_coef * ce + b_coef * ces;
  }
}

// ---------------------------------------------------------------------------
extern "C" void kernel_launch(void* const* d_in, const int* in_sizes, int n_in,
                              void* d_out, int out_size, void* d_ws, size_t ws_size,
                              hipStream_t stream) {
  (void)n_in; (void)out_size; (void)ws_size;
  const float* x   = (const float*)d_in[0];
  const int*   tgt = (const int*)d_in[1];
  const int Brows = in_sizes[1];
  const int C     = in_sizes[0] / Brows;

  float* ce_arr = (float*)d_ws;          // Brows floats
  float* pt_arr = ce_arr + Brows;        // Brows floats
  float* out    = (float*)d_out;

  const int grid = 2048;
  if (C == C_FIX) {
    const size_t shmem = (size_t)2 * BUF_F4 * sizeof(float4);  // 81920 B
    (void)hipFuncSetAttribute(reinterpret_cast<const void*>(ces_rows_fast),
                              hipFuncAttributeMaxDynamicSharedMemorySize, (int)shmem);
    ces_rows_fast<<<grid, TPB, shmem, stream>>>(x, tgt, ce_arr, pt_arr, Brows);
  } else {
    ces_rows_generic<<<grid, 256, 0, stream>>>(x, tgt, ce_arr, pt_arr, Brows, C);
  }
  ces_finalize<<<1, 256, 0, stream>>>(ce_arr, pt_arr, out, Brows);
}